// logBarrierLoss_78176994722349
// MI455X (gfx1250) — compile-verified
//
#include <hip/hip_runtime.h>

typedef __attribute__((ext_vector_type(2))) float v2f;
typedef __attribute__((ext_vector_type(4))) float v4f;
typedef __attribute__((ext_vector_type(8))) float v8f;

#define LOW_BAND  131050.0f
#define HIGH_BAND 131100.0f

constexpr int BATCH = 64;
constexpr int PIX   = 512 * 512;      // pixels per image per channel
constexpr int SPLIT = 32;             // blocks per image
constexpr int CHUNK = PIX / SPLIT;    // 8192 pixels per block
constexpr int TPB   = 256;            // 8 waves (wave32)
constexpr int VITER = CHUNK / (TPB * 4); // 8 float4 loads per thread per channel

// Stage 1: per-block partial sums of sigmoid(x1 - x0).
__global__ __launch_bounds__(TPB)
void logbarrier_partial_kernel(const float* __restrict__ prob,
                               float* __restrict__ ws) {
    const int b = blockIdx.x / SPLIT;       // image
    const int c = blockIdx.x % SPLIT;       // chunk within image
    const float* __restrict__ p0 = prob + (size_t)b * (2 * PIX) + (size_t)c * CHUNK;
    const float* __restrict__ p1 = p0 + PIX;

    const int t = threadIdx.x;
    float acc = 0.0f;

#pragma unroll
    for (int i = 0; i < VITER; ++i) {
        const int off = i * (TPB * 4) + t * 4;
        v4f x0 = __builtin_nontemporal_load((const v4f*)(p0 + off));
        v4f x1 = __builtin_nontemporal_load((const v4f*)(p1 + off));
#pragma unroll
        for (int j = 0; j < 4; ++j) {
            // sigmoid(x1 - x0) = 1 / (1 + exp(x0 - x1))
            float e = __expf(x0[j] - x1[j]);
            acc += __builtin_amdgcn_rcpf(1.0f + e);
        }
    }

    // --- wave32 reduction via V_WMMA_F32_16X16X4_F32 (full f32 precision) ---
    // A (16x4 f32, 2 VGPRs): lanes 0-15 hold K=0, lanes 16-31 hold K=2 in VGPR0.
    // With B = ones(4x16): D[m][n] = acc[lane m] + acc[lane m+16].
    v2f a;  a.x = acc;  a.y = 0.0f;
    v2f bb; bb.x = 1.0f; bb.y = 1.0f;
    v8f cz = {};
    v8f d = __builtin_amdgcn_wmma_f32_16x16x4_f32(
        /*neg_a=*/false, a, /*neg_b=*/false, bb,
        /*c_mod=*/(short)0, cz, /*reuse_a=*/false, /*reuse_b=*/false);

    // Per lane: sum the 8 C/D VGPRs (rows m = lanehalf*8 .. +7), then fold halves.
    float w = d[0] + d[1] + d[2] + d[3] + d[4] + d[5] + d[6] + d[7];
    w += __shfl_xor(w, 16, 32);

    __shared__ float lds[TPB / 32];
    const int lane = t & 31, wave = t >> 5;
    if (lane == 0) lds[wave] = w;
    __syncthreads();
    if (t == 0) {
        float tot = 0.0f;
#pragma unroll
        for (int i = 0; i < TPB / 32; ++i) tot += lds[i];
        ws[blockIdx.x] = tot;
    }
}

// Stage 2: assemble s[b], then reproduce the reference epilogue exactly.
__global__ __launch_bounds__(BATCH)
void logbarrier_final_kernel(const float* __restrict__ ws,
                             float* __restrict__ out) {
    __shared__ float s_sh[BATCH];
    const int b = threadIdx.x;
    float s = 0.0f;
#pragma unroll
    for (int c = 0; c < SPLIT; ++c) s += ws[b * SPLIT + c];
    s_sh[b] = s;
    __syncthreads();

    if (b == 0) {
        float lowsum = 0.0f;
        for (int i = 0; i < BATCH; ++i) {
            float dd = s_sh[i] - LOW_BAND;
            lowsum += dd * dd;
        }
        float csum = 0.0f, cnt = 0.0f;
        for (int i = 0; i < BATCH; ++i) {
            float si = s_sh[i];
            if (si >= HIGH_BAND) {            // checked first, as in the reference
                float dd = si - HIGH_BAND;
                csum += dd * dd; cnt += 1.0f;
            } else if (si <= LOW_BAND) {
                csum += lowsum; cnt += (float)BATCH;
            } else {
                cnt += 1.0f;
            }
        }
        out[0] = csum / cnt;
    }
}

extern "C" void kernel_launch(void* const* d_in, const int* in_sizes, int n_in,
                              void* d_out, int out_size, void* d_ws, size_t ws_size,
                              hipStream_t stream) {
    (void)in_sizes; (void)n_in; (void)out_size; (void)ws_size;
    const float* prob = (const float*)d_in[0];   // [64, 2, 512, 512] f32
    float* ws  = (float*)d_ws;                    // 2048 f32 partials (8 KB)
    float* out = (float*)d_out;                   // 1 f32

    logbarrier_partial_kernel<<<BATCH * SPLIT, TPB, 0, stream>>>(prob, ws);
    logbarrier_final_kernel<<<1, BATCH, 0, stream>>>(ws, out);
}